// Frames2Results_84722524881316
// MI455X (gfx1250) — compile-verified
//
#include <hip/hip_runtime.h>
#include <hip/hip_bf16.h>
#include <cstdint>

#define DEVFN __device__ __forceinline__

namespace {

constexpr int B = 32, C = 2, MAXN = 100, NL = 5;
constexpr int HWA[NL]  = {36864, 9216, 2304, 576, 144};
constexpr int WA[NL]   = {192, 96, 48, 24, 12};
constexpr int STRA[NL] = {8, 16, 32, 64, 128};
constexpr int LOFF[NL] = {0, 1000, 2000, 3000, 3576};
constexpr int M  = 3720;          // candidates per image after top-k (1000+1000+1000+576+144)
constexpr int MC = M * C;
constexpr float RCLIP     = 4.135166556742356f;  // log(1000/16)
constexpr float IMGW      = 1536.0f;
constexpr float SCORE_THR = 0.05f;
constexpr float IOU_THR   = 0.6f;

// ---- workspace layout (float units) ----
constexpr int KOFF[3] = {0, B * HWA[0], B * (HWA[0] + HWA[1])};
constexpr int KEYTOT  = B * (HWA[0] + HWA[1] + HWA[2]);   // 1,548,288 floats
constexpr int CBOX    = KEYTOT;                           // B*M*4 floats
constexpr int CSCORE  = CBOX + B * M * 4;                 // B*M*2 floats

// ---- NMS LDS layout (bytes within dynamic shared; needs CDNA5 320KB-LDS WGP) ----
constexpr int SM_SC    = M * 16;            // boxes: [0, 59520); scores: [59520, 89280)
constexpr int SM_RV    = SM_SC + MC * 4;
constexpr int SM_RI    = SM_RV + 128;
constexpr int SM_SELV  = SM_RI + 128;
constexpr int SM_SELI  = SM_SELV + 4;
constexpr int SMEM_BYTES = ((SM_SELI + 4 + 127) / 128) * 128;   // ~89.7 KB

struct Ptrs { const float* cls[NL]; const float* bbox[NL]; const float* ctr[NL]; };

DEVFN float sigm(float x) { return 1.0f / (1.0f + expf(-x)); }
DEVFN float clampf(float x, float lo, float hi) { return fminf(fmaxf(x, lo), hi); }

// ---- CDNA5 async global->LDS (ASYNCcnt-tracked), 16B per lane ----
DEVFN void async_g2l_b128(uint32_t lds_byte, const void* gbase, uint32_t gbyte) {
  asm volatile("global_load_async_to_lds_b128 %0, %1, %2"
               :: "v"(lds_byte), "v"(gbyte), "s"(gbase)
               : "memory");
}
DEVFN void wait_async0() {
  asm volatile("s_wait_asynccnt 0" ::: "memory");
}
// LDS aperture flat address: its low 32 bits are the LDS offset (ISA §10.2)
DEVFN uint32_t lds_addr_of(const void* p) { return (uint32_t)(uintptr_t)p; }

// Decode one location into candidate arrays (box + per-class thresholded scores).
// Fully specialized per level: hw/w/stride/loff are immediates.
template <int LVL>
DEVFN void emit_cand(const Ptrs& P, float* __restrict__ ws, int b, int p, int slot) {
  constexpr int hw = HWA[LVL];
  constexpr int w  = WA[LVL];
  constexpr float s = (float)STRA[LVL];
  const float* cl = P.cls[LVL]  + (size_t)b * C * hw;
  const float* bb = P.bbox[LVL] + (size_t)b * 4 * hw;
  const float* ct = P.ctr[LVL]  + (size_t)b * hw;

  float c  = sigm(ct[p]);
  float s0 = sigm(cl[p]) * c;
  float s1 = sigm(cl[hw + p]) * c;

  float dx = bb[p];
  float dy = bb[hw + p];
  float dw = clampf(bb[2 * hw + p], -RCLIP, RCLIP);
  float dh = clampf(bb[3 * hw + p], -RCLIP, RCLIP);

  int y = p / w, x = p - y * w;          // constant divisor -> mul/shift
  constexpr float aw = 8.0f * (float)STRA[LVL];   // anchor side = OCTAVE_SCALE*stride
  float cx  = (float)x * s, cy = (float)y * s;    // anchor center = grid*stride
  float pcx = cx + dx * aw, pcy = cy + dy * aw;
  float pwh = 0.5f * aw * expf(dw);
  float phh = 0.5f * aw * expf(dh);
  float x1 = clampf(pcx - pwh, 0.0f, IMGW);
  float y1 = clampf(pcy - phh, 0.0f, IMGW);
  float x2 = clampf(pcx + pwh, 0.0f, IMGW);
  float y2 = clampf(pcy + phh, 0.0f, IMGW);

  int ci = b * M + LOFF[LVL] + slot;
  float* cbx = ws + CBOX + (size_t)ci * 4;
  cbx[0] = x1; cbx[1] = y1; cbx[2] = x2; cbx[3] = y2;
  float* cs = ws + CSCORE + (size_t)ci * 2;
  cs[0] = (s0 > SCORE_THR) ? s0 : -1.0f;
  cs[1] = (s1 > SCORE_THR) ? s1 : -1.0f;
}

// ---- Kernel 1 (x3): ranking key = max_c sigmoid(cls) * sigmoid(ctr), big levels ----
template <int LVL>
__global__ void key_kernel(Ptrs P, float* __restrict__ ws) {
  constexpr int hw = HWA[LVL];
  int idx = blockIdx.x * blockDim.x + threadIdx.x;
  if (idx >= B * hw) return;
  int b = idx / hw, p = idx - b * hw;    // constant divisor
  const float* cl = P.cls[LVL] + (size_t)b * C * hw;
  const float* ct = P.ctr[LVL] + (size_t)b * hw;
  __builtin_prefetch(cl + p + 2048, 0, 0);      // global_prefetch_b8 (speculative)
  float s0 = sigm(cl[p]);
  float s1 = sigm(cl[hw + p]);
  float c  = sigm(ct[p]);
  ws[KOFF[LVL] + idx] = fmaxf(s0, s1) * c;
}

// ---- Kernel 2 (x2): small levels (k == hw): decode everything directly ----
template <int LVL>
__global__ void small_kernel(Ptrs P, float* __restrict__ ws) {
  constexpr int hw = HWA[LVL];
  int idx = blockIdx.x * blockDim.x + threadIdx.x;
  if (idx >= B * hw) return;
  int b = idx / hw, p = idx - b * hw;
  emit_cand<LVL>(P, ws, b, p, p);
}

// ---- Kernel 3 (x3): LDS-histogram top-k (k=1000) per image ----
template <int LVL>
__global__ __launch_bounds__(1024) void select_kernel(Ptrs P, float* __restrict__ ws) {
  __shared__ unsigned hist[2048];
  __shared__ int sT, sNeed, selCnt, tieCnt;
  constexpr int hw = HWA[LVL];
  constexpr int K  = 1000;
  const int b   = blockIdx.x;
  const int tid = threadIdx.x;
  const float* key = ws + KOFF[LVL] + b * hw;

  for (int i = tid; i < 2048; i += 1024) hist[i] = 0u;
  if (tid == 0) { selCnt = 0; tieCnt = 0; }
  __syncthreads();

  for (int p = tid; p < hw; p += 1024) {
    int bin = min((int)(key[p] * 2048.0f), 2047);
    atomicAdd(&hist[bin], 1u);
  }
  __syncthreads();

  if (tid == 0) {
    int cum = 0, t = 0, above = 0; bool found = false;
    for (int i = 2047; i >= 0; --i) {
      int cnt = (int)hist[i];
      if (cum + cnt >= K) { t = i; above = cum; found = true; break; }
      cum += cnt;
    }
    if (!found) { t = 0; above = cum; }
    sT = t; sNeed = K - above;
  }
  __syncthreads();

  const int T = sT, need = sNeed;
  for (int p = tid; p < hw; p += 1024) {
    int bin = min((int)(key[p] * 2048.0f), 2047);
    bool take = (bin > T);
    if (!take && bin == T) take = (atomicAdd(&tieCnt, 1) < need);
    if (take) {
      int slot = atomicAdd(&selCnt, 1);   // exactly K slots filled
      emit_cand<LVL>(P, ws, b, p, slot);
    }
  }
}

// ---- Kernel 4: greedy class-aware NMS, fully LDS-resident ----
__global__ __launch_bounds__(1024) void nms_kernel(const float* __restrict__ ws,
                                                   float* __restrict__ out) {
  extern __shared__ char smem[];
  float* sbox   = (float*)smem;              // M*4 floats
  float* sscore = (float*)(smem + SM_SC);    // MC floats
  float* rval   = (float*)(smem + SM_RV);
  int*   ridx   = (int*)  (smem + SM_RI);
  float* sselv  = (float*)(smem + SM_SELV);
  int*   sseli  = (int*)  (smem + SM_SELI);

  const int b = blockIdx.x, tid = threadIdx.x;
  const char* gb = (const char*)(ws + CBOX   + (size_t)b * M * 4);
  const char* gs = (const char*)(ws + CSCORE + (size_t)b * MC);
  const uint32_t ldsBox = lds_addr_of(sbox);
  const uint32_t ldsSc  = lds_addr_of(sscore);

  // Stage candidates into LDS via CDNA5 async global->LDS path (16B/lane).
  for (uint32_t off = (uint32_t)tid * 16u; off < (uint32_t)(M * 16); off += 1024u * 16u)
    async_g2l_b128(ldsBox + off, gb, off);
  for (uint32_t off = (uint32_t)tid * 16u; off < (uint32_t)(MC * 4); off += 1024u * 16u)
    async_g2l_b128(ldsSc + off, gs, off);
  wait_async0();
  __builtin_amdgcn_s_wait_tensorcnt(0);
  __syncthreads();

  for (int it = 0; it < MAXN; ++it) {
    // --- parallel argmax over MC scores (first-occurrence tie-break) ---
    float bv = -1e30f; int bi = MC;
    for (int j = tid; j < MC; j += 1024) {
      float v = sscore[j];
      if (v > bv) { bv = v; bi = j; }
    }
    for (int o = 16; o > 0; o >>= 1) {          // wave32 shuffle reduce
      float ov = __shfl_down(bv, o, 32);
      int   oi = __shfl_down(bi, o, 32);
      if (ov > bv || (ov == bv && oi < bi)) { bv = ov; bi = oi; }
    }
    if ((tid & 31) == 0) { rval[tid >> 5] = bv; ridx[tid >> 5] = bi; }
    __syncthreads();
    if (tid < 32) {
      bv = rval[tid]; bi = ridx[tid];
      for (int o = 16; o > 0; o >>= 1) {
        float ov = __shfl_down(bv, o, 32);
        int   oi = __shfl_down(bi, o, 32);
        if (ov > bv || (ov == bv && oi < bi)) { bv = ov; bi = oi; }
      }
      if (tid == 0) { *sselv = bv; *sseli = bi; }
    }
    __syncthreads();

    const float val = *sselv;
    const int j = *sseli;
    const int m = j >> 1, cls = j & 1;

    if (tid == 0) {
      float* dr = out + ((size_t)b * MAXN + it) * 5;
      float* lr = out + (size_t)B * MAXN * 5 + (size_t)b * MAXN + it;
      if (val > 0.0f) {
        dr[0] = sbox[m * 4 + 0]; dr[1] = sbox[m * 4 + 1];
        dr[2] = sbox[m * 4 + 2]; dr[3] = sbox[m * 4 + 3];
        dr[4] = val; *lr = (float)cls;
      } else {
        dr[0] = dr[1] = dr[2] = dr[3] = dr[4] = 0.0f; *lr = -1.0f;
      }
    }

    // Class-offset NMS == same-class suppression (offset=(W+1)*label makes
    // cross-class IoU exactly 0). Self-IoU=1 retires the winner itself.
    if (val > 0.0f) {
      const float x1 = sbox[m * 4 + 0], y1 = sbox[m * 4 + 1];
      const float x2 = sbox[m * 4 + 2], y2 = sbox[m * 4 + 3];
      const float a  = (x2 - x1) * (y2 - y1);
      for (int m2 = tid; m2 < M; m2 += 1024) {
        float bx1 = sbox[m2 * 4 + 0], by1 = sbox[m2 * 4 + 1];
        float bx2 = sbox[m2 * 4 + 2], by2 = sbox[m2 * 4 + 3];
        float ix1 = fmaxf(x1, bx1), iy1 = fmaxf(y1, by1);
        float ix2 = fminf(x2, bx2), iy2 = fminf(y2, by2);
        float inter = fmaxf(ix2 - ix1, 0.0f) * fmaxf(iy2 - iy1, 0.0f);
        float a2 = (bx2 - bx1) * (by2 - by1);
        float iou = inter / fmaxf(a + a2 - inter, 1e-6f);
        if (iou > IOU_THR) sscore[m2 * 2 + cls] = -1.0f;
      }
    }
    __syncthreads();
  }
}

}  // namespace

extern "C" void kernel_launch(void* const* d_in, const int* in_sizes, int n_in,
                              void* d_out, int out_size, void* d_ws, size_t ws_size,
                              hipStream_t stream) {
  (void)in_sizes; (void)n_in; (void)out_size; (void)ws_size;
  Ptrs P;
  for (int l = 0; l < NL; ++l) {
    P.cls[l]  = (const float*)d_in[3 * l + 0];
    P.bbox[l] = (const float*)d_in[3 * l + 1];
    P.ctr[l]  = (const float*)d_in[3 * l + 2];
  }
  float* ws  = (float*)d_ws;
  float* out = (float*)d_out;

  key_kernel<0><<<(B * HWA[0] + 255) / 256, 256, 0, stream>>>(P, ws);
  key_kernel<1><<<(B * HWA[1] + 255) / 256, 256, 0, stream>>>(P, ws);
  key_kernel<2><<<(B * HWA[2] + 255) / 256, 256, 0, stream>>>(P, ws);
  small_kernel<3><<<(B * HWA[3] + 255) / 256, 256, 0, stream>>>(P, ws);
  small_kernel<4><<<(B * HWA[4] + 255) / 256, 256, 0, stream>>>(P, ws);
  select_kernel<0><<<B, 1024, 0, stream>>>(P, ws);
  select_kernel<1><<<B, 1024, 0, stream>>>(P, ws);
  select_kernel<2><<<B, 1024, 0, stream>>>(P, ws);
  nms_kernel<<<B, 1024, SMEM_BYTES, stream>>>(ws, out);
}